// PSDPeakDetectorEncoder_37039797960744
// MI455X (gfx1250) — compile-verified
//
#include <hip/hip_runtime.h>

// PSD peak detector: per-row argmax over [16384, 4096] f32, affine map to BPM,
// broadcast to [16384, 512]. Pure streaming problem: 256 MiB read + 32 MiB
// write => ~12.4 us floor at 23.3 TB/s. One wave32 per row; NT cache hints so
// the one-pass 256 MiB input doesn't thrash the 192 MB L2.

#define N_BINS 4096
#define HIDDEN 512
#define BATCH  16384

// Native clang vector type: required by __builtin_nontemporal_load/store
// (HIP's float4 is a class and is rejected).
typedef float v4f __attribute__((ext_vector_type(4)));

__global__ __launch_bounds__(256) void psd_peak_kernel(const float* __restrict__ x,
                                                       float* __restrict__ out) {
    const int gtid = blockIdx.x * 256 + threadIdx.x;
    const int row  = gtid >> 5;           // one wave (32 lanes) per row
    const int lane = threadIdx.x & 31;
    if (row >= BATCH) return;

    const v4f* row4 = reinterpret_cast<const v4f*>(x) + (size_t)row * (N_BINS / 4);

    // Per-lane scan in increasing index order; strict '>' keeps first max.
    float bestV = -3.402823466e+38f;
    int   bestI = 0;

    #pragma unroll 8
    for (int k = 0; k < (N_BINS / 4) / 32; ++k) {       // 32 iterations
        const int v4i = k * 32 + lane;                   // coalesced float4 id
        const v4f v = __builtin_nontemporal_load(row4 + v4i);
        const int b = v4i * 4;
        if (v.x > bestV) { bestV = v.x; bestI = b;     }
        if (v.y > bestV) { bestV = v.y; bestI = b + 1; }
        if (v.z > bestV) { bestV = v.z; bestI = b + 2; }
        if (v.w > bestV) { bestV = v.w; bestI = b + 3; }
    }

    // Wave32 butterfly reduction: (max value, smallest index on tie) is an
    // associative+commutative semilattice -> exact jnp.argmax semantics.
    #pragma unroll
    for (int m = 16; m >= 1; m >>= 1) {
        const float ov = __shfl_xor(bestV, m, 32);
        const int   oi = __shfl_xor(bestI, m, 32);
        if (ov > bestV || (ov == bestV && oi < bestI)) { bestV = ov; bestI = oi; }
    }

    // Match reference fp32 op order: FMIN + (FMAX-FMIN)*idx/(F-1), then *60.
    const float freq = 0.1f + (0.4f * (float)bestI) / (float)(N_BINS - 1);
    const float bpm  = freq * 60.0f;

    // Broadcast-store 512 floats for this row: 4 x b128 per lane, coalesced,
    // non-temporal (write-once output).
    v4f* o4 = reinterpret_cast<v4f*>(out) + (size_t)row * (HIDDEN / 4);
    v4f ov4;
    ov4.x = bpm; ov4.y = bpm; ov4.z = bpm; ov4.w = bpm;
    #pragma unroll
    for (int j = 0; j < (HIDDEN / 4) / 32; ++j) {        // 4 iterations
        __builtin_nontemporal_store(ov4, o4 + j * 32 + lane);
    }
}

extern "C" void kernel_launch(void* const* d_in, const int* in_sizes, int n_in,
                              void* d_out, int out_size, void* d_ws, size_t ws_size,
                              hipStream_t stream) {
    const float* x  = (const float*)d_in[0];
    float* out      = (float*)d_out;

    // 8 rows per 256-thread block (1 wave32 per row) -> 2048 blocks.
    const int rows_per_block = 256 / 32;
    const int grid = BATCH / rows_per_block;
    psd_peak_kernel<<<grid, 256, 0, stream>>>(x, out);
}